// EB_19490561589325
// MI455X (gfx1250) — compile-verified
//
#include <hip/hip_runtime.h>
#include <hip/hip_bf16.h>

// ---- problem constants (from reference) ----
#define B_SZ   4
#define NPTS   4096
#define DFEAT  8
#define NNBR   16
#define DM     16
#define DO     8
#define NM     64
#define PNM    64
#define NROWS  15          // N_NBR - 1 rows per point per order
#define ROWS_PER_BATCH (NPTS * NROWS)   // 61440
#define PTS_PER_BLK 8
#define ROWS_PER_BLK (PTS_PER_BLK * NROWS)  // 120 (padded to 128)
#define BLKS_PER_BATCH (NPTS / PTS_PER_BLK) // 512
#define BIGV 99999999.0f

typedef float v2f __attribute__((ext_vector_type(2)));
typedef float v8f __attribute__((ext_vector_type(8)));

__device__ __forceinline__ v8f wmma_f32(v2f a, v2f b, v8f c) {
    // V_WMMA_F32_16X16X4_F32 : D = A(16x4) * B(4x16) + C(16x16), all fp32
    return __builtin_amdgcn_wmma_f32_16x16x4_f32(
        false, a, false, b, (short)0, c, false, false);
}

// ---------------------------------------------------------------------------
// K0: zx[b][k] = z[b]@Wmz + bmz ; zv[b][m] = z[b]@Wvz + bvz
// ---------------------------------------------------------------------------
__global__ void EB_k0_zproj(const float* __restrict__ z,
                            const float* __restrict__ Wmz, const float* __restrict__ bmz,
                            const float* __restrict__ Wvz, const float* __restrict__ bvz,
                            float* __restrict__ zx, float* __restrict__ zv) {
    int tid = threadIdx.x;
    if (tid < B_SZ * DO) {
        int b = tid >> 3, k = tid & 7;
        float acc = bmz[k];
        #pragma unroll
        for (int c = 0; c < PNM; ++c) acc += z[b * PNM + c] * Wmz[c * DO + k];
        zx[b * DO + k] = acc;
    }
    if (tid < B_SZ * NM) {
        int b = tid >> 6, m = tid & 63;
        float acc = bvz[m];
        #pragma unroll
        for (int c = 0; c < PNM; ++c) acc += z[b * PNM + c] * Wvz[c * NM + m];
        zv[b * NM + m] = acc;
    }
}

// ---------------------------------------------------------------------------
// K1: 16-nearest-neighbor indices per point (one block per (b, i) row)
// ---------------------------------------------------------------------------
__global__ void EB_k1_knn(const float* __restrict__ x, int* __restrict__ idx_out) {
    __shared__ float d2row[NPTS];
    __shared__ float redv[256];
    __shared__ int   redi[256];

    const int i = blockIdx.x;
    const int b = blockIdx.y;
    const int tid = threadIdx.x;

    const float* xb = x + (size_t)b * NPTS * DFEAT;
    float xi[DFEAT];
    #pragma unroll
    for (int d = 0; d < DFEAT; ++d) xi[d] = xb[(size_t)i * DFEAT + d];

    // squared distances (monotonic with sqrt); self & zero-distance -> BIG
    for (int s = 0; s < NPTS / 256; ++s) {
        int j = s * 256 + tid;
        const float* xj = xb + (size_t)j * DFEAT;
        float d2 = 0.f;
        #pragma unroll
        for (int d = 0; d < DFEAT; ++d) { float t = xi[d] - xj[d]; d2 += t * t; }
        d2row[j] = (j == i || d2 <= 0.f) ? BIGV : d2;
    }
    __syncthreads();

    for (int r = 0; r < NNBR; ++r) {
        float bv = BIGV * 2.f; int bi = NPTS;
        for (int s = 0; s < NPTS / 256; ++s) {
            int j = s * 256 + tid;
            float v = d2row[j];
            if (v < bv) { bv = v; bi = j; }
        }
        redv[tid] = bv; redi[tid] = bi;
        __syncthreads();
        for (int off = 128; off > 0; off >>= 1) {
            if (tid < off) {
                float ov = redv[tid + off]; int oi = redi[tid + off];
                if (ov < redv[tid] || (ov == redv[tid] && oi < redi[tid])) {
                    redv[tid] = ov; redi[tid] = oi;
                }
            }
            __syncthreads();
        }
        if (tid == 0) {
            idx_out[((size_t)b * NPTS + i) * NNBR + r] = redi[0];
            d2row[redi[0]] = BIGV;
        }
        __syncthreads();
    }
}

// ---------------------------------------------------------------------------
// K2: fused order-features + WMMA mixing GEMMs + relu + means
//     grid = (512, B), block = 256 (8 waves); each block owns 8 points.
// ---------------------------------------------------------------------------
__global__ void EB_k2_main(const float* __restrict__ x, const int* __restrict__ idx,
                           const float* __restrict__ Wm2, const float* __restrict__ bm2,
                           const float* __restrict__ Wm3, const float* __restrict__ bm3,
                           const float* __restrict__ Wv2, const float* __restrict__ bv2,
                           const float* __restrict__ Wv3, const float* __restrict__ bv3,
                           const float* __restrict__ Wmx, const float* __restrict__ bmx,
                           const float* __restrict__ Wvx, const float* __restrict__ bvx,
                           const float* __restrict__ zx, const float* __restrict__ zv,
                           float* __restrict__ out_x, float* __restrict__ z_part) {
    __shared__ float s_wmx[32 * 16];  // Wmx padded N 8->16
    __shared__ float s_wvx[32 * 64];
    __shared__ float s_bmx[16];
    __shared__ float s_bvx[64];
    __shared__ float s_zx[8];
    __shared__ float s_zv[64];
    __shared__ float s_wm2[32], s_bm2[16], s_wm3[48], s_bm3[16];
    __shared__ float s_wv2[32], s_bv2[16], s_wv3[48], s_bv3[16];
    // phase 1/2: moments [128][32] + v_moments [128][32] (8192 f)
    // phase 3/4: xr [128][8] (1024 f) + zr [128][64] (8192 f) = 9216 f
    __shared__ float s_buf[9216];

    const int tid = threadIdx.x;
    const int b   = blockIdx.y;
    const int p0  = blockIdx.x * PTS_PER_BLK;

    // ---- phase 0: stage weights / broadcast vectors ----
    for (int i = tid; i < 32 * 16; i += 256) {
        int k = i >> 4, n = i & 15;
        s_wmx[i] = (n < DO) ? Wmx[k * DO + n] : 0.f;
    }
    for (int i = tid; i < 32 * 64; i += 256) s_wvx[i] = Wvx[i];
    if (tid < 16) s_bmx[tid] = (tid < DO) ? bmx[tid] : 0.f;
    if (tid < 64) s_bvx[tid] = bvx[tid];
    if (tid < 8)  s_zx[tid]  = zx[b * DO + tid];
    if (tid < 64) s_zv[tid]  = zv[b * NM + tid];
    if (tid < 32) { s_wm2[tid] = Wm2[tid]; s_wv2[tid] = Wv2[tid]; }
    if (tid < 48) { s_wm3[tid] = Wm3[tid]; s_wv3[tid] = Wv3[tid]; }
    if (tid < 16) { s_bm2[tid] = bm2[tid]; s_bm3[tid] = bm3[tid];
                    s_bv2[tid] = bv2[tid]; s_bv3[tid] = bv3[tid]; }
    __syncthreads();

    // ---- phase 1: moment rows (half row per thread) ----
    {
        int row = tid >> 1;          // 0..127 (120 valid)
        int h   = tid & 1;           // 0: order-2 cols 0..15, 1: order-3 cols 16..31
        int pl  = row / NROWS;
        int t   = row - pl * NROWS;
        if (row >= ROWS_PER_BLK) { pl = PTS_PER_BLK - 1; t = 0; }
        const int n_pt = p0 + pl;
        const int* nb = idx + ((size_t)b * NPTS + n_pt) * NNBR;
        const float* xb = x + (size_t)b * NPTS * DFEAT;

        if (h == 0) {
            int j0 = nb[0], j1 = nb[t + 1];
            float g0[DFEAT], g1[DFEAT];
            #pragma unroll
            for (int d = 0; d < DFEAT; ++d) {
                g0[d] = xb[(size_t)j0 * DFEAT + d];
                g1[d] = xb[(size_t)j1 * DFEAT + d];
            }
            #pragma unroll
            for (int k = 0; k < DM; ++k) {
                float am = 0.f, av = 0.f;
                float wm0 = s_wm2[k], wm1 = s_wm2[16 + k], bm = s_bm2[k];
                float wv0 = s_wv2[k], wv1 = s_wv2[16 + k], bv = s_bv2[k];
                #pragma unroll
                for (int d = 0; d < DFEAT; ++d) {
                    am += fmaxf(g0[d] * wm0 + g1[d] * wm1 + bm, 0.f);
                    av += fmaxf(g0[d] * wv0 + g1[d] * wv1 + bv, 0.f);
                }
                s_buf[row * 32 + k]        = am * 0.125f;
                s_buf[4096 + row * 32 + k] = av * 0.125f;
            }
        } else {
            // ts3: (0, 1, t+2) for t<14, (0, 2, 3) for t==14
            int a  = (t < 14) ? 1 : 2;
            int c3 = (t < 14) ? (t + 2) : 3;
            int j0 = nb[0], ja = nb[a], jc = nb[c3];
            float g0[DFEAT], ga[DFEAT], gc[DFEAT];
            #pragma unroll
            for (int d = 0; d < DFEAT; ++d) {
                g0[d] = xb[(size_t)j0 * DFEAT + d];
                ga[d] = xb[(size_t)ja * DFEAT + d];
                gc[d] = xb[(size_t)jc * DFEAT + d];
            }
            #pragma unroll
            for (int k = 0; k < DM; ++k) {
                float am = 0.f, av = 0.f;
                float wm0 = s_wm3[k], wm1 = s_wm3[16 + k], wm2_ = s_wm3[32 + k], bm = s_bm3[k];
                float wv0 = s_wv3[k], wv1 = s_wv3[16 + k], wv2_ = s_wv3[32 + k], bv = s_bv3[k];
                #pragma unroll
                for (int d = 0; d < DFEAT; ++d) {
                    am += fmaxf(g0[d] * wm0 + ga[d] * wm1 + gc[d] * wm2_ + bm, 0.f);
                    av += fmaxf(g0[d] * wv0 + ga[d] * wv1 + gc[d] * wv2_ + bv, 0.f);
                }
                s_buf[row * 32 + 16 + k]        = am * 0.125f;
                s_buf[4096 + row * 32 + 16 + k] = av * 0.125f;
            }
        }
    }
    __syncthreads();

    // ---- phase 2: load fp32 WMMA A-fragments (16x4 tiles, K=32 -> 8 steps) ----
    const int wave  = tid >> 5;
    const int lane  = tid & 31;
    const int mrow0 = wave << 4;
    const int mloc  = lane & 15;
    const int khalf = (lane >> 4) << 1;   // lanes 16..31 hold K+2, K+3
    v2f am[8], av[8];
    #pragma unroll
    for (int kk = 0; kk < 8; ++kk) {
        int kb = kk * 4 + khalf;
        am[kk].x = s_buf[(mrow0 + mloc) * 32 + kb];
        am[kk].y = s_buf[(mrow0 + mloc) * 32 + kb + 1];
        av[kk].x = s_buf[4096 + (mrow0 + mloc) * 32 + kb];
        av[kk].y = s_buf[4096 + (mrow0 + mloc) * 32 + kb + 1];
    }
    __syncthreads();   // s_buf is repurposed below

    // ---- phase 3: WMMA GEMMs + bias + z broadcast + relu ----
    const int col = lane & 15;
    // x path: 16x16x32 via 8 WMMA (cols 8..15 are zero padding)
    {
        v8f cx = {0.f,0.f,0.f,0.f,0.f,0.f,0.f,0.f};
        #pragma unroll
        for (int kk = 0; kk < 8; ++kk) {
            int kb = kk * 4 + khalf;
            v2f bw; bw.x = s_wmx[kb * 16 + col]; bw.y = s_wmx[(kb + 1) * 16 + col];
            cx = wmma_f32(am[kk], bw, cx);
        }
        #pragma unroll
        for (int r = 0; r < 8; ++r) {
            int row = mrow0 + r + 8 * (lane >> 4);
            if (col < DO) {
                float v = fmaxf(cx[r] + s_bmx[col] + s_zx[col], 0.f);
                s_buf[row * DO + col] = v;    // xr region [0,1024)
            }
        }
    }
    // z path: N=64 -> 4 tiles of 16, 8 WMMA each
    #pragma unroll
    for (int tz = 0; tz < 4; ++tz) {
        v8f cz = {0.f,0.f,0.f,0.f,0.f,0.f,0.f,0.f};
        #pragma unroll
        for (int kk = 0; kk < 8; ++kk) {
            int kb = kk * 4 + khalf;
            v2f bw; bw.x = s_wvx[kb * 64 + tz * 16 + col];
                    bw.y = s_wvx[(kb + 1) * 64 + tz * 16 + col];
            cz = wmma_f32(av[kk], bw, cz);
        }
        int colg = tz * 16 + col;
        #pragma unroll
        for (int r = 0; r < 8; ++r) {
            int row = mrow0 + r + 8 * (lane >> 4);
            float v = fmaxf(cz[r] + s_bvx[colg] + s_zv[colg], 0.f);
            s_buf[1024 + row * 64 + colg] = v;  // zr region
        }
    }
    __syncthreads();

    // ---- phase 4: deterministic reductions ----
    if (tid < 64) {
        // x output: mean over 15 rows per point
        int p = tid >> 3, k = tid & 7;
        float acc = 0.f;
        #pragma unroll
        for (int tr = 0; tr < NROWS; ++tr) acc += s_buf[(p * NROWS + tr) * DO + k];
        out_x[((size_t)b * NPTS + (p0 + p)) * DO + k] = acc * (1.0f / NROWS);
        // z partial: sum of 120 valid rows for this column
        float zacc = 0.f;
        for (int row = 0; row < ROWS_PER_BLK; ++row) zacc += s_buf[1024 + row * 64 + tid];
        z_part[((size_t)b * BLKS_PER_BATCH + blockIdx.x) * NM + tid] = zacc;
    }
}

// ---------------------------------------------------------------------------
// K3: reduce z partials -> z_new
// ---------------------------------------------------------------------------
__global__ void EB_k3_zfinal(const float* __restrict__ z_part, float* __restrict__ out_z) {
    int tid = threadIdx.x;            // 256 = B*NM
    int b = tid >> 6, m = tid & 63;
    float acc = 0.f;
    for (int blk = 0; blk < BLKS_PER_BATCH; ++blk)
        acc += z_part[((size_t)b * BLKS_PER_BATCH + blk) * NM + m];
    out_z[tid] = acc * (1.0f / (float)ROWS_PER_BATCH);
}

// ---------------------------------------------------------------------------
extern "C" void kernel_launch(void* const* d_in, const int* in_sizes, int n_in,
                              void* d_out, int out_size, void* d_ws, size_t ws_size,
                              hipStream_t stream) {
    const float* x   = (const float*)d_in[0];
    const float* z   = (const float*)d_in[1];
    const float* Wm2 = (const float*)d_in[2];  const float* bm2 = (const float*)d_in[3];
    const float* Wm3 = (const float*)d_in[4];  const float* bm3 = (const float*)d_in[5];
    const float* Wv2 = (const float*)d_in[6];  const float* bv2 = (const float*)d_in[7];
    const float* Wv3 = (const float*)d_in[8];  const float* bv3 = (const float*)d_in[9];
    const float* Wmx = (const float*)d_in[10]; const float* bmx = (const float*)d_in[11];
    const float* Wvx = (const float*)d_in[12]; const float* bvx = (const float*)d_in[13];
    const float* Wmz = (const float*)d_in[14]; const float* bmz = (const float*)d_in[15];
    const float* Wvz = (const float*)d_in[16]; const float* bvz = (const float*)d_in[17];

    char* ws = (char*)d_ws;
    int*   idx    = (int*)  (ws);                                   // B*NP*16 ints (1 MB)
    float* zx     = (float*)(ws + (size_t)B_SZ * NPTS * NNBR * 4);  // B*8
    float* zv     = zx + B_SZ * DO;                                 // B*64
    float* z_part = zv + B_SZ * NM;                                 // B*512*64 (512 KB)

    float* out_x = (float*)d_out;                       // B*NP*DO = 131072
    float* out_z = out_x + (size_t)B_SZ * NPTS * DO;    // B*NM   = 256

    EB_k0_zproj<<<1, 256, 0, stream>>>(z, Wmz, bmz, Wvz, bvz, zx, zv);
    EB_k1_knn<<<dim3(NPTS, B_SZ), 256, 0, stream>>>(x, idx);
    EB_k2_main<<<dim3(BLKS_PER_BATCH, B_SZ), 256, 0, stream>>>(
        x, idx, Wm2, bm2, Wm3, bm3, Wv2, bv2, Wv3, bv3,
        Wmx, bmx, Wvx, bvx, zx, zv, out_x, z_part);
    EB_k3_zfinal<<<1, 256, 0, stream>>>(z_part, out_z);
}